// ContourIntegrationLayer_23029614641784
// MI455X (gfx1250) — compile-verified
//
#include <hip/hip_runtime.h>
#include <stdint.h>

typedef float f4 __attribute__((ext_vector_type(4)));

namespace {
constexpr int Hc = 56, Wc = 56, Cc = 256;
constexpr int THREADS       = 256;                       // 8 waves (wave32)
constexpr int V4_PER_THREAD = 8;
constexpr int V4_PER_BLOCK  = THREADS * V4_PER_THREAD;   // 2048
constexpr int TOTAL_V4      = 32 * Hc * Wc * (Cc / 4);   // 6,422,528
constexpr int NBLOCKS       = TOTAL_V4 / V4_PER_BLOCK;   // 3136 (exact)
}

__global__ __launch_bounds__(THREADS)
void contour3x3_dw(const float* __restrict__ x,
                   const float* __restrict__ kw,
                   float* __restrict__ out)
{
  __shared__ __align__(16) float ldsk[9 * Cc];   // 9 taps x 256 channels = 9216 B

  const int tid = threadIdx.x;

  // ---- Stage filter taps into LDS via gfx1250 async DMA (ASYNCcnt path) ----
  // 576 b128 chunks; lanes 0..191 take exactly 3 each (wave-uniform guard:
  // waves 6,7 skip entirely; no exec-masked loop).
  if (tid < 192) {
    const uint32_t lbase = (uint32_t)(uintptr_t)&ldsk[0];
    const uint32_t base  = (uint32_t)tid * 48u;          // 3 x 16 B per lane
    #pragma unroll
    for (int j = 0; j < 3; ++j) {
      const uint32_t off = base + (uint32_t)j * 16u;
      asm volatile("global_load_async_to_lds_b128 %0, %1, %2"
                   :: "v"(lbase + off), "v"(off), "s"(kw)
                   : "memory");
    }
  }
  asm volatile("s_wait_asynccnt 0x0" ::: "memory");      // wave's own async ops done
  __syncthreads();                                       // publish LDS to all waves

  // Channel group of this lane is invariant across its 8 outputs (256 % 64 == 0):
  // hoist the 8 non-center taps into VGPRs once (8 x ds_load_b128).
  const int cb = (tid & 63) << 2;
  f4 kt[8];
  {
    int j = 0;
    #pragma unroll
    for (int t = 0; t < 9; ++t) {
      if (t == 4) continue;                              // center tap excluded
      kt[j++] = *(const f4*)&ldsk[t * Cc + cb];
    }
  }

  const int base_v4 = blockIdx.x * V4_PER_BLOCK + tid;

  #pragma unroll
  for (int it = 0; it < V4_PER_THREAD; ++it) {
    const int v   = base_v4 + it * THREADS;
    const int pix = v >> 6;                              // / (C/4)
    const int w   = pix % Wc;
    const int h   = (pix / Wc) % Hc;

    const float* xc = x + ((size_t)v << 2);
    f4 acc = *(const f4*)xc;                             // residual (center tap is zero)

    // All 32 lanes of a wave share one pixel -> this branch is wave-uniform.
    if ((unsigned)(h - 1) < (unsigned)(Hc - 2) &&
        (unsigned)(w - 1) < (unsigned)(Wc - 2)) {
      // Interior fast path (92.9% of pixels): 8 unconditional b128 loads off one
      // base address (imm offsets within +/-58368 B < 2^23), then 32 FMAs.
      #pragma unroll
      for (int t = 0; t < 8; ++t) {
        const int tap = (t < 4) ? t : t + 1;             // skip center (tap 4)
        const int dy  = tap / 3 - 1;
        const int dx  = tap % 3 - 1;
        const f4 xv = *(const f4*)(xc + (dy * Wc + dx) * Cc);
        acc = acc + xv * kt[t];
      }
    } else {
      // Border path: SAME padding -> out-of-range taps contribute zero.
      int j = 0;
      #pragma unroll
      for (int dy = -1; dy <= 1; ++dy) {
        #pragma unroll
        for (int dx = -1; dx <= 1; ++dx) {
          if (dy == 0 && dx == 0) continue;
          const f4 kv = kt[j++];
          if ((unsigned)(h + dy) < (unsigned)Hc &&
              (unsigned)(w + dx) < (unsigned)Wc) {
            const f4 xv = *(const f4*)(xc + (dy * Wc + dx) * Cc);
            acc = acc + xv * kv;
          }
        }
      }
    }

    // Streaming store: keep L2 for the (reused, fully L2-resident) input tensor.
    __builtin_nontemporal_store(acc, (f4*)(out + ((size_t)v << 2)));
  }
}

extern "C" void kernel_launch(void* const* d_in, const int* in_sizes, int n_in,
                              void* d_out, int out_size, void* d_ws, size_t ws_size,
                              hipStream_t stream) {
  (void)in_sizes; (void)n_in; (void)out_size; (void)d_ws; (void)ws_size;
  const float* x  = (const float*)d_in[0];   // [32,56,56,256] f32
  const float* kw = (const float*)d_in[1];   // [3,3,256] f32
  float* out      = (float*)d_out;           // [32,56,56,256] f32
  hipLaunchKernelGGL(contour3x3_dw, dim3(NBLOCKS), dim3(THREADS), 0, stream,
                     x, kw, out);
}